// EMDLoss_28063316312808
// MI455X (gfx1250) — compile-verified
//
#include <hip/hip_runtime.h>
#include <hip/hip_bf16.h>
#include <cfloat>

#define BATCH 8
#define NPTS  2048

typedef __attribute__((ext_vector_type(2))) float v2f;
typedef __attribute__((ext_vector_type(8))) float v8f;

// ---------------------------------------------------------------------------
// Kernel 1: D[b][i][j] = |x1_i|^2 + |x2_j|^2 - 2 * dot(x1_i, x2_j)
// Cross term via V_WMMA_F32_16X16X4_F32 (A: 16x4 rows of xyz1 with K=(x,y,z,0),
// B: 4x16 cols of xyz2). One 16x16 tile per wave, 8 waves per 256-thread block.
// ---------------------------------------------------------------------------
__global__ __launch_bounds__(256) void emd_dist_wmma(const float* __restrict__ xyz1,
                                                     const float* __restrict__ xyz2,
                                                     float* __restrict__ D) {
  const int lane = threadIdx.x & 31;
  const int wave = threadIdx.x >> 5;
  const int b    = blockIdx.y;
  const int tile = blockIdx.x * 8 + wave;   // 16384 tiles per batch (128 x 128)
  const int ti   = tile >> 7;
  const int tj   = tile & 127;
  const int i0   = ti * 16;
  const int j0   = tj * 16;

  const float* x1 = xyz1 + (size_t)b * NPTS * 3;
  const float* x2 = xyz2 + (size_t)b * NPTS * 3;
  float*       Db = D    + (size_t)b * NPTS * NPTS;

  const int  m  = lane & 15;
  const bool hi = lane >= 16;

  // A-matrix 16x4 f32 layout: lanes 0-15 hold (K=0,K=1), lanes 16-31 hold (K=2,K=3)
  const int arow = i0 + m;
  v2f a;
  if (!hi) { a.x = x1[arow * 3 + 0]; a.y = x1[arow * 3 + 1]; }
  else     { a.x = x1[arow * 3 + 2]; a.y = 0.0f;             }

  // B-matrix 4x16 f32 layout (mirror of A): lanes 0-15 -> rows K=0,1; lanes 16-31 -> K=2,3
  const int bcol = j0 + m;
  v2f bm;
  if (!hi) { bm.x = x2[bcol * 3 + 0]; bm.y = x2[bcol * 3 + 1]; }
  else     { bm.x = x2[bcol * 3 + 2]; bm.y = 0.0f;             }

  v8f c = {};
  // (neg_a, A, neg_b, B, c_mod, C, reuse_a, reuse_b)
  c = __builtin_amdgcn_wmma_f32_16x16x4_f32(false, a, false, bm, (short)0, c,
                                            false, false);

  // |x2_j|^2 for this lane's column
  const float cx = x2[bcol * 3 + 0];
  const float cy = x2[bcol * 3 + 1];
  const float cz = x2[bcol * 3 + 2];
  const float ncol = cx * cx + cy * cy + cz * cz;

  // C/D layout: lane -> column j0+(lane&15); VGPR r -> row r (+8 for lanes>=16)
#pragma unroll
  for (int r = 0; r < 8; ++r) {
    const int   row  = i0 + r + (hi ? 8 : 0);
    const float rx   = x1[row * 3 + 0];
    const float ry   = x1[row * 3 + 1];
    const float rz   = x1[row * 3 + 2];
    const float nrow = rx * rx + ry * ry + rz * rz;
    Db[(size_t)row * NPTS + bcol] = nrow + ncol - 2.0f * c[r];
  }
}

// ---------------------------------------------------------------------------
// Kernel 2: sequential greedy matching, one 1024-thread workgroup per batch.
// Per step: 2 cols/thread -> wave32 shfl argmin -> 32-entry cross-wave reduce.
// Tie-break prefers lowest column index (matches jnp.argmin).
// ---------------------------------------------------------------------------
__global__ __launch_bounds__(1024) void emd_greedy(const float* __restrict__ xyz1,
                                                   const float* __restrict__ xyz2,
                                                   const float* __restrict__ D,
                                                   float* __restrict__ results,
                                                   int useD) {
  __shared__ float    sx1x[NPTS], sx1y[NPTS], sx1z[NPTS];
  __shared__ float    sx2x[NPTS], sx2y[NPTS], sx2z[NPTS];
  __shared__ unsigned used[NPTS / 32];
  __shared__ float    wred_v[32];
  __shared__ int      wred_i[32];
  __shared__ float    s_vmin;
  __shared__ int      s_jmin;

  const int t    = threadIdx.x;
  const int lane = t & 31;
  const int wav  = t >> 5;
  const int b    = blockIdx.x;

  const float* x1 = xyz1 + (size_t)b * NPTS * 3;
  const float* x2 = xyz2 + (size_t)b * NPTS * 3;
  const float* Db = D    + (size_t)b * NPTS * NPTS;

  for (int j = t; j < NPTS; j += 1024) {
    sx1x[j] = x1[j * 3 + 0]; sx1y[j] = x1[j * 3 + 1]; sx1z[j] = x1[j * 3 + 2];
    sx2x[j] = x2[j * 3 + 0]; sx2y[j] = x2[j * 3 + 1]; sx2z[j] = x2[j * 3 + 2];
  }
  if (t < NPTS / 32) used[t] = 0u;
  __syncthreads();

  float sum = 0.0f;  // meaningful in thread 0 only
  for (int i = 0; i < NPTS; ++i) {
    const float px = sx1x[i], py = sx1y[i], pz = sx1z[i];
    const float* Drow = Db + (size_t)i * NPTS;

    float best = FLT_MAX;
    int   bidx = 0;
#pragma unroll
    for (int k = 0; k < 2; ++k) {
      const int j = t + k * 1024;
      float v;
      if (useD) {
        v = Drow[j];
      } else {
        const float dx = px - sx2x[j];
        const float dy = py - sx2y[j];
        const float dz = pz - sx2z[j];
        v = dx * dx + dy * dy + dz * dz;
      }
      const bool  u  = (used[j >> 5] >> (j & 31)) & 1u;
      const float mv = u ? FLT_MAX : v;
      if (mv < best) { best = mv; bidx = j; }  // strict '<' keeps lower j on tie
    }

    // wave32 argmin reduce (index tie-break)
#pragma unroll
    for (int off = 16; off > 0; off >>= 1) {
      const float ov = __shfl_xor(best, off, 32);
      const int   oi = __shfl_xor(bidx, off, 32);
      if (ov < best || (ov == best && oi < bidx)) { best = ov; bidx = oi; }
    }
    if (lane == 0) { wred_v[wav] = best; wred_i[wav] = bidx; }
    __syncthreads();

    if (t < 32) {
      float v   = wred_v[t];
      int   idx = wred_i[t];
#pragma unroll
      for (int off = 16; off > 0; off >>= 1) {
        const float ov = __shfl_xor(v, off, 32);
        const int   oi = __shfl_xor(idx, off, 32);
        if (ov < v || (ov == v && oi < idx)) { v = ov; idx = oi; }
      }
      if (t == 0) { s_vmin = v; s_jmin = idx; }
    }
    __syncthreads();

    if (t == 0) {
      sum += s_vmin;  // winner is unmasked, so masked value == raw row value
      const int jm = s_jmin;
      used[jm >> 5] |= (1u << (jm & 31));
    }
    __syncthreads();
  }

  if (t == 0) results[b] = sum / (float)NPTS;
}

// ---------------------------------------------------------------------------
// Kernel 3: deterministic final reduce (no float atomics across replays)
// ---------------------------------------------------------------------------
__global__ void emd_finalize(const float* __restrict__ results,
                             float* __restrict__ out) {
  float s = 0.0f;
  for (int b = 0; b < BATCH; ++b) s += results[b];
  out[0] = s / (float)BATCH;
}

extern "C" void kernel_launch(void* const* d_in, const int* in_sizes, int n_in,
                              void* d_out, int out_size, void* d_ws, size_t ws_size,
                              hipStream_t stream) {
  const float* xyz1 = (const float*)d_in[0];
  const float* xyz2 = (const float*)d_in[1];

  const size_t D_elems = (size_t)BATCH * NPTS * NPTS;
  const bool   useD    = ws_size >= (D_elems + BATCH) * sizeof(float);

  float* Dptr = (float*)d_ws;
  float* res  = useD ? ((float*)d_ws + D_elems) : (float*)d_ws;

  if (useD) {
    // 16384 tiles/batch, 8 waves (tiles) per 256-thread block -> 2048 blocks/batch
    emd_dist_wmma<<<dim3(2048, BATCH), 256, 0, stream>>>(xyz1, xyz2, Dptr);
  }
  emd_greedy<<<BATCH, 1024, 0, stream>>>(xyz1, xyz2, Dptr, res, useD ? 1 : 0);
  emd_finalize<<<1, 1, 0, stream>>>(res, (float*)d_out);
}